// GraphSAGE_24618752541196
// MI455X (gfx1250) — compile-verified
//
#include <hip/hip_runtime.h>

// CDNA5 / gfx1250, wave32.
typedef __attribute__((ext_vector_type(2))) float v2f;
typedef __attribute__((ext_vector_type(8))) float v8f;

static constexpr int DIM = 64;   // D == H == 64

// ---------------------------------------------------------------- zero fill
__global__ __launch_bounds__(256)
void zero_f32_kernel(float* __restrict__ p, long long n) {
  long long i = (long long)blockIdx.x * blockDim.x + threadIdx.x;
  long long stride = (long long)gridDim.x * blockDim.x;
  for (; i < n; i += stride) p[i] = 0.0f;
}

// ------------------------------------------- weighted scatter (segment sum)
// One wave32 per edge: lanes load 2 consecutive floats of h[src] (coalesced
// 256B), scale by edge weight, atomic-add into num[dst]. h (25.6MB) is
// L2-resident on MI455X (192MB L2), so this is L2 atomic-throughput bound.
__global__ __launch_bounds__(256)
void sage_scatter_kernel(const float* __restrict__ h,
                         const int*  __restrict__ src,
                         const int*  __restrict__ dst,
                         const float* __restrict__ ew,
                         float* __restrict__ num,
                         float* __restrict__ den,
                         int E) {
  int e    = (int)(((unsigned)blockIdx.x * blockDim.x + threadIdx.x) >> 5);
  int lane = threadIdx.x & 31;
  if (e >= E) return;
  int   s = src[e];
  int   d = dst[e];
  float w = ew[e];
  float2 v = ((const float2*)(h + (size_t)s * DIM))[lane];
  float* np = num + (size_t)d * DIM + lane * 2;
  atomicAdd(np,     v.x * w);
  atomicAdd(np + 1, v.y * w);
  if (lane == 0) atomicAdd(den + d, w);
}

// ----------------------------------------------------- fused SAGE dense part
// hout[16x64 tile] = (num/max(den,1e-6)) @ Wl^T + bias + hin @ Wr^T
// One wave per 16 output rows; four 16x16 N-tiles accumulated with
// V_WMMA_F32_16X16X4_F32 (fp32 WMMA — matches reference precision).
//   A 16x4 layout: lanes 0-15 hold (M=lane, K=k,k+1), lanes 16-31 (K=k+2,k+3)
//   B 4x16 layout: VGPR0 = K {k+half}, VGPR1 = K {k+2+half}, N = lane&15
//   C 16x16 layout: VGPR j -> M = j + 8*half, N = lane&15
// Weights+bias staged in LDS (32.25 KB of the 320 KB/WGP); row indices are
// CLAMPED (not predicated) so all hot-loop loads are unconditional — tail
// lanes recompute row n-1's identical output, stores are benign duplicates.
__global__ __launch_bounds__(256)
void sage_gemm_kernel(const float* __restrict__ num,
                      const float* __restrict__ den,
                      const float* __restrict__ hin,
                      const float* __restrict__ Wl,    // [64][64] row-major (out,in)
                      const float* __restrict__ bias,  // [64]
                      const float* __restrict__ Wr,    // [64][64]
                      float* __restrict__ hout,
                      int n) {
  __shared__ float sWl[DIM * DIM];
  __shared__ float sWr[DIM * DIM];
  __shared__ float sb[DIM];

  // cooperative stage: 2 x 16 KB weights + bias, float4 vectorized
  for (int i = threadIdx.x; i < DIM * DIM / 4; i += blockDim.x) {
    ((float4*)sWl)[i] = ((const float4*)Wl)[i];
    ((float4*)sWr)[i] = ((const float4*)Wr)[i];
  }
  if (threadIdx.x < DIM) sb[threadIdx.x] = bias[threadIdx.x];
  __syncthreads();

  const int lane = threadIdx.x & 31;
  const int half = lane >> 4;    // 0 -> K pair {0,1}, 1 -> K pair {2,3}
  const int mrow = lane & 15;
  int tile = blockIdx.x * (blockDim.x >> 5) + (threadIdx.x >> 5);
  int row_base = tile * 16;
  if (row_base >= n) return;

  int row = min(row_base + mrow, n - 1);          // clamp: no exec branches
  float rd = 1.0f / fmaxf(den[row], 1e-6f);
  const float* arow = num + (size_t)row * DIM;
  const float* xrow = hin + (size_t)row * DIM;

  v8f acc[4] = {v8f{}, v8f{}, v8f{}, v8f{}};

  // agg @ Wl^T   (B[kk][nc] = Wl[nc][kk])
  #pragma unroll
  for (int k = 0; k < DIM; k += 4) {
    int kk = k + half * 2;
    float2 av = *(const float2*)(arow + kk);      // one global_load_b64
    v2f a;  a.x = av.x * rd;  a.y = av.y * rd;
    #pragma unroll
    for (int nt = 0; nt < 4; ++nt) {
      int nc = nt * 16 + mrow;
      v2f b;
      b.x = sWl[nc * DIM + (k + half)];
      b.y = sWl[nc * DIM + (k + 2 + half)];
      acc[nt] = __builtin_amdgcn_wmma_f32_16x16x4_f32(
          false, a, false, b, (short)0, acc[nt], false, false);
    }
  }
  // + hin @ Wr^T
  #pragma unroll
  for (int k = 0; k < DIM; k += 4) {
    int kk = k + half * 2;
    float2 av = *(const float2*)(xrow + kk);
    v2f a;  a.x = av.x;  a.y = av.y;
    #pragma unroll
    for (int nt = 0; nt < 4; ++nt) {
      int nc = nt * 16 + mrow;
      v2f b;
      b.x = sWr[nc * DIM + (k + half)];
      b.y = sWr[nc * DIM + (k + 2 + half)];
      acc[nt] = __builtin_amdgcn_wmma_f32_16x16x4_f32(
          false, a, false, b, (short)0, acc[nt], false, false);
    }
  }

  // + bias, store (clamped rows rewrite row n-1 with identical values)
  #pragma unroll
  for (int nt = 0; nt < 4; ++nt) {
    int nc = nt * 16 + mrow;
    float bv = sb[nc];
    #pragma unroll
    for (int j = 0; j < 8; ++j) {
      int r = min(row_base + j + half * 8, n - 1);
      hout[(size_t)r * DIM + nc] = acc[nt][j] + bv;
    }
  }
}

// ------------------------------------------------------------ edge predictor
// One wave32 per supervision edge; 2 features per lane; wave32 shfl reduction.
__global__ __launch_bounds__(256)
void edge_predict_kernel(const float* __restrict__ h,
                         const int*  __restrict__ si,
                         const int*  __restrict__ sj,
                         const float* __restrict__ wwp, const float* __restrict__ bwp,
                         const float* __restrict__ wep, const float* __restrict__ bep,
                         float* __restrict__ out, int ES) {
  int e    = (int)(((unsigned)blockIdx.x * blockDim.x + threadIdx.x) >> 5);
  int lane = threadIdx.x & 31;
  if (e >= ES) return;
  int i = si[e], j = sj[e];
  float2 hi = ((const float2*)(h + (size_t)i * DIM))[lane];
  float2 hj = ((const float2*)(h + (size_t)j * DIM))[lane];
  float2 s = {hi.x + hj.x, hi.y + hj.y};
  float2 p = {hi.x * hj.x, hi.y * hj.y};
  float2 ww0 = ((const float2*)wwp)[lane];
  float2 ww1 = ((const float2*)(wwp + DIM))[lane];
  float2 we0 = ((const float2*)wep)[lane];
  float2 we1 = ((const float2*)(wep + DIM))[lane];
  float aw = s.x * ww0.x + s.y * ww0.y + p.x * ww1.x + p.y * ww1.y;
  float ae = s.x * we0.x + s.y * we0.y + p.x * we1.x + p.y * we1.y;
  #pragma unroll
  for (int off = 16; off >= 1; off >>= 1) {
    aw += __shfl_xor(aw, off, 32);
    ae += __shfl_xor(ae, off, 32);
  }
  if (lane == 0) {
    out[e]      = fmaxf(aw + bwp[0], 0.0f);  // edge_weights (relu)
    out[ES + e] = ae + bep[0];               // edge_predictor
  }
}

// ---------------------------------------------------------------------------
extern "C" void kernel_launch(void* const* d_in, const int* in_sizes, int n_in,
                              void* d_out, int out_size, void* d_ws, size_t ws_size,
                              hipStream_t stream) {
  const float* x    = (const float*)d_in[0];
  const int*   sup  = (const int*)  d_in[1];   // [2, ES]
  const int*   medg = (const int*)  d_in[2];   // [2, E]
  const float* mew  = (const float*)d_in[3];
  const float* w1l  = (const float*)d_in[4];
  const float* b1   = (const float*)d_in[5];
  const float* w1r  = (const float*)d_in[6];
  const float* w2l  = (const float*)d_in[7];
  const float* b2   = (const float*)d_in[8];
  const float* w2r  = (const float*)d_in[9];
  const float* wep  = (const float*)d_in[10];
  const float* bep  = (const float*)d_in[11];
  const float* wwp  = (const float*)d_in[12];
  const float* bwp  = (const float*)d_in[13];

  const int N  = in_sizes[0] / DIM;
  const int E  = in_sizes[3];
  const int ES = in_sizes[1] / 2;

  // workspace: num[N*64] | den[N] | h1[N*64] | h2[N*64]  (~77 MB)
  float* num = (float*)d_ws;
  float* den = num + (size_t)N * DIM;
  float* h1  = den + N;
  float* h2  = h1 + (size_t)N * DIM;

  const int* msrc = medg;
  const int* mdst = medg + E;
  const int* ssrc = sup;
  const int* sdst = sup + ES;

  const long long nz = (long long)N * DIM + N;  // num+den contiguous
  const int scat_blocks = (int)(((long long)E * 32 + 255) / 256);
  const int tiles       = (N + 15) / 16;
  const int gemm_blocks = (tiles + 7) / 8;      // 8 waves per 256-thread block
  const int pred_blocks = (int)(((long long)ES * 32 + 255) / 256);

  // layer 1
  zero_f32_kernel<<<2048, 256, 0, stream>>>(num, nz);
  sage_scatter_kernel<<<scat_blocks, 256, 0, stream>>>(x, msrc, mdst, mew, num, den, E);
  sage_gemm_kernel<<<gemm_blocks, 256, 0, stream>>>(num, den, x, w1l, b1, w1r, h1, N);

  // layer 2
  zero_f32_kernel<<<2048, 256, 0, stream>>>(num, nz);
  sage_scatter_kernel<<<scat_blocks, 256, 0, stream>>>(h1, msrc, mdst, mew, num, den, E);
  sage_gemm_kernel<<<gemm_blocks, 256, 0, stream>>>(num, den, h1, w2l, b2, w2r, h2, N);

  // edge predictor
  edge_predict_kernel<<<pred_blocks, 256, 0, stream>>>(h2, ssrc, sdst, wwp, bwp,
                                                       wep, bep, (float*)d_out, ES);
}